// SeparableMonteCarloMaxPoolingV2_15247133900972
// MI455X (gfx1250) — compile-verified
//
#include <hip/hip_runtime.h>
#include <stdint.h>

// Problem constants (from reference setup_inputs)
#define B_  16
#define N_  65536
#define P_  64
#define M_  4096
#define L_  9
#define NP_ (N_ * P_)            // 4,194,304 floats per batch slice (16 MB)
#define MP_ (M_ * P_)            // 262,144 (m,p) pairs
#define PAIRS_PER_BLOCK 256
#define IDX_PER_BLOCK   (PAIRS_PER_BLOCK * L_)   // 2304 int32 = 9216 bytes

typedef __attribute__((ext_vector_type(4))) unsigned int uint32x4;
typedef __attribute__((ext_vector_type(8))) int          int32x8;
typedef __attribute__((ext_vector_type(4))) int          int32x4;

__global__ __launch_bounds__(PAIRS_PER_BLOCK)
void smcmp_max_kernel(const float* __restrict__ x,
                      const int*   __restrict__ lrf,
                      float*       __restrict__ out) {
    __shared__ int s_idx[IDX_PER_BLOCK];

    const int tid = threadIdx.x;
    const int f0  = blockIdx.x * PAIRS_PER_BLOCK;   // first flat (m*64+p) pair of block

    // ---- Stage this block's 2304 contiguous indices into LDS ----
#if __has_builtin(__builtin_amdgcn_tensor_load_to_lds) && __has_builtin(__builtin_amdgcn_s_wait_tensorcnt)
    // TDM path: wave 0 issues one 1D tensor DMA (2304 x 4B), waits TENSORcnt.
    const int wave = __builtin_amdgcn_readfirstlane(tid >> 5);
    if (wave == 0) {
        const uint64_t gaddr   = (uint64_t)(uintptr_t)(lrf + (size_t)f0 * L_);
        const uint32_t lds_off = (uint32_t)(uintptr_t)(&s_idx[0]);   // addr[31:0] == LDS offset

        uint32x4 g0;
        g0[0] = 1u;                                        // count=1, user mode, no gather
        g0[1] = lds_off;                                   // lds_addr
        g0[2] = (uint32_t)gaddr;                           // global_addr[31:0]
        g0[3] = (uint32_t)((gaddr >> 32) & 0x1FFFFFFu)     // global_addr[56:32]
              | (2u << 30);                                // type = 2 ("image")

        const uint32_t td0   = IDX_PER_BLOCK;              // tensor_dim0 (elements)
        const uint32_t tile0 = IDX_PER_BLOCK;              // tile_dim0  (elements)
        int32x8 g1;
        g1[0] = (int)0x00020000u;                          // data_size=2 (4B), wg_mask=0, no flags
        g1[1] = (int)((td0 & 0xFFFFu) << 16);              // tensor_dim0[15:0] @ bits 63:48
        g1[2] = (int)((td0 >> 16) | (1u << 16));           // tensor_dim0[31:16]; tensor_dim1=1
        g1[3] = (int)(tile0 << 16);                        // tensor_dim1[31:16]=0; tile_dim0
        g1[4] = 0;                                         // tile_dim1=0, tile_dim2=0 -> 1D tile
        g1[5] = (int)td0;                                  // tensor_dim0_stride[31:0]
        g1[6] = 0;                                         // stride0[47:32]=0, stride1[15:0]=0
        g1[7] = 0;

        const int32x4 z4 = {0, 0, 0, 0};
        const int32x8 z8 = {0, 0, 0, 0, 0, 0, 0, 0};
        __builtin_amdgcn_tensor_load_to_lds(g0, g1, z4, z4, z8, 0);
        __builtin_amdgcn_s_wait_tensorcnt((short)0);
    }
#else
    // Fallback: cooperative coalesced copy.
    for (int i = tid; i < IDX_PER_BLOCK; i += PAIRS_PER_BLOCK)
        s_idx[i] = lrf[(size_t)f0 * L_ + i];
#endif
    __syncthreads();

    // ---- Convert this thread's 9 indices to byte offsets within one batch slice ----
    // offset = (idx*P + p)*4 = idx*256 + p*4  (max 16,777,212 -> fits u32: GVS addressing)
    const uint32_t p4 = ((uint32_t)(f0 + tid) & (P_ - 1u)) * 4u;
    uint32_t boff[L_];
#pragma unroll
    for (int l = 0; l < L_; ++l) {
        const uint32_t idx = (uint32_t)s_idx[tid * L_ + l];
        boff[l] = idx * (P_ * 4u) + p4;
    }

    const char* __restrict__ xbytes = (const char*)x;
    float*      __restrict__ op     = out + (f0 + tid);

    // ---- 16 batches x 9 gathers each; indices/offsets reused across all batches ----
#pragma unroll
    for (int b = 0; b < B_; ++b) {
        const char* base = xbytes + (size_t)b * ((size_t)NP_ * 4u);
        float v0 = *(const float*)(base + boff[0]);
        float v1 = *(const float*)(base + boff[1]);
        float v2 = *(const float*)(base + boff[2]);
        float v3 = *(const float*)(base + boff[3]);
        float v4 = *(const float*)(base + boff[4]);
        float v5 = *(const float*)(base + boff[5]);
        float v6 = *(const float*)(base + boff[6]);
        float v7 = *(const float*)(base + boff[7]);
        float v8 = *(const float*)(base + boff[8]);
        // Balanced 9-way max -> v_max3_num_f32 friendly
        const float m0 = fmaxf(fmaxf(v0, v1), v2);
        const float m1 = fmaxf(fmaxf(v3, v4), v5);
        const float m2 = fmaxf(fmaxf(v6, v7), v8);
        const float r  = fmaxf(fmaxf(m0, m1), m2);
        // Streaming store: don't pollute L2 (gather working set lives there)
        __builtin_nontemporal_store(r, op + (size_t)b * MP_);
    }
}

extern "C" void kernel_launch(void* const* d_in, const int* in_sizes, int n_in,
                              void* d_out, int out_size, void* d_ws, size_t ws_size,
                              hipStream_t stream) {
    const float* x   = (const float*)d_in[0];   // (16, 65536, 64) f32
    const int*   lrf = (const int*)d_in[1];     // (4096, 64, 9) i32
    float*       out = (float*)d_out;           // (16, 4096, 64) f32

    dim3 grid(MP_ / PAIRS_PER_BLOCK);           // 1024 blocks
    dim3 block(PAIRS_PER_BLOCK);                // 256 threads = 8 wave32
    hipLaunchKernelGGL(smcmp_max_kernel, grid, block, 0, stream, x, lrf, out);
}